// DelayedSynapticLayer_43155831390397
// MI455X (gfx1250) — compile-verified
//
#include <hip/hip_runtime.h>
#include <hip/hip_bf16.h>
#include <stdint.h>

// Problem constants (DelayedSynapticLayer): buf[B,D,P], weight[P,N], delay_raw[P,N]
#define B_SZ   32
#define D_LEN  51        // D_MAX + 1
#define D_PAD  52        // pad so taps (d, d+1) always in-bounds & bank-friendly
#define P_SZ   1024
#define N_SZ   1024

#define NB        128                 // threads per block: one n-column per thread
#define N_BLOCKS  (N_SZ / NB)         // 8
#define P_BLOCKS  32                  // p-split across grid.y (deterministic partials)
#define P_PER_BLK (P_SZ / P_BLOCKS)   // 32
#define PT        8                   // p-tile staged in LDS (8*52*32*4 = 53,248 B)
#define TILES     (P_PER_BLK / PT)    // 4
#define ROW_STEP  (NB / PT)           // 16 rows advanced per staging iteration
#define STAGE_ITERS ((B_SZ * D_LEN * PT) / NB)   // 102

// Builtin signature: (int addrspace(1)* gsrc, int addrspace(3)* ldst, imm off, imm cpol)
typedef __attribute__((address_space(1))) int* gas1_i32;
typedef __attribute__((address_space(3))) int* las3_i32;

__device__ __forceinline__ void async_copy_b32(const float* g, float* l) {
#if __has_builtin(__builtin_amdgcn_global_load_async_to_lds_b32)
  // async DMA of one dword into LDS; tracked by ASYNCcnt (CDNA5 path)
  __builtin_amdgcn_global_load_async_to_lds_b32((gas1_i32)g, (las3_i32)l, 0, 0);
#else
  *l = *g;  // synchronous fallback (still correct)
#endif
}

__device__ __forceinline__ void wait_async0() {
#if __has_builtin(__builtin_amdgcn_s_wait_asynccnt)
  __builtin_amdgcn_s_wait_asynccnt(0);
#elif __has_builtin(__builtin_amdgcn_global_load_async_to_lds_b32)
  asm volatile("s_wait_asynccnt 0" ::: "memory");
#endif
}

__global__ __launch_bounds__(NB)
void dsl_main_kernel(const float* __restrict__ buf,
                     const float* __restrict__ weight,
                     const float* __restrict__ delay_raw,
                     float* __restrict__ partial) {
  // LDS tile layout: lds[(b*PT + j)*D_PAD + d]  (taps d,d+1 adjacent so they fuse
  // into one 2-address DS op; D_PAD=52 keeps distinct df values in distinct banks)
  __shared__ float lds[B_SZ * PT * D_PAD];

  const int tid  = threadIdx.x;
  const int nblk = blockIdx.x;       // [0, N_BLOCKS)
  const int pblk = blockIdx.y;       // [0, P_BLOCKS)
  const int n      = nblk * NB + tid;
  const int p_base = pblk * P_PER_BLK;

  float acc[B_SZ];
#pragma unroll
  for (int b = 0; b < B_SZ; ++b) acc[b] = 0.0f;

  for (int t = 0; t < TILES; ++t) {
    const int p0 = p_base + t * PT;

    // ---- stage: buf[b, d, p0+j] -> lds[(b*PT+j)*D_PAD + d], async to LDS ----
    // Per-thread: fixed j = tid%8; row = b*51 + d starts at tid/8 and advances
    // by 16 each iteration. Global index is simply row*1024 + p0 + j (constant
    // pointer stride); (b, d) advance incrementally with a single wrap check.
    {
      const int j = tid & (PT - 1);
      int d = tid >> 3;              // initial row < 16 => b = 0, d = row
      int b = 0;
      const float* gp = buf + ((tid >> 3) << 10) + p0 + j;
#pragma unroll 1
      for (int k = 0; k < STAGE_ITERS; ++k) {
        async_copy_b32(gp, &lds[(b * PT + j) * D_PAD + d]);
        gp += (ROW_STEP << 10);      // +16 rows in global memory
        d += ROW_STEP;
        const int wrap = (d >= D_LEN);
        d -= wrap ? D_LEN : 0;
        b += wrap;
      }
    }
    // zero the d = 51 pad slot (read when df == 50, weighted by alpha == 0)
    for (int e = tid; e < B_SZ * PT; e += NB)
      lds[e * D_PAD + D_LEN] = 0.0f;

    wait_async0();
    __syncthreads();

    // ---- compute: this thread's column n over PT pre-neurons ----
#pragma unroll 1
    for (int jp = 0; jp < PT; ++jp) {
      const int p = p0 + jp;
      const float x = delay_raw[p * N_SZ + n];
      const float w = weight[p * N_SZ + n];
      // d_cont = 50 * sigmoid(x) in (0, 50]; floor in [0, 50]
      const float sg    = 1.0f / (1.0f + __expf(-x));
      const float dcont = 50.0f * sg;
      const float dff   = floorf(dcont);
      const int   df    = (int)dff;
      const float alpha = dcont - dff;
      const float wc    = w * alpha;
      const float wf    = w - wc;           // w * (1 - alpha)
      const int   base  = jp * D_PAD + df;
#pragma unroll
      for (int b = 0; b < B_SZ; ++b) {
        const int idx = b * (PT * D_PAD) + base;
        const float sf = lds[idx];          // tap at df
        const float sc = lds[idx + 1];      // tap at df + 1 (pad-safe)
        acc[b] = fmaf(wf, sf, fmaf(wc, sc, acc[b]));
      }
    }
    __syncthreads();   // protect LDS before next tile overwrites it
  }

  // deterministic partials: partial[pblk][b][n]
#pragma unroll
  for (int b = 0; b < B_SZ; ++b)
    partial[((pblk * B_SZ + b) << 10) + n] = acc[b];
}

__global__ __launch_bounds__(256)
void dsl_reduce_kernel(const float* __restrict__ partial,
                       float* __restrict__ out) {
  const int i = blockIdx.x * blockDim.x + threadIdx.x;   // over B*N
  if (i >= B_SZ * N_SZ) return;
  const int b = i >> 10;
  const int n = i & (N_SZ - 1);
  float s = 0.0f;
#pragma unroll
  for (int k = 0; k < P_BLOCKS; ++k)
    s += partial[((k * B_SZ + b) << 10) + n];
  out[i] = s;   // I_syn[b, n]
}

extern "C" void kernel_launch(void* const* d_in, const int* in_sizes, int n_in,
                              void* d_out, int out_size, void* d_ws, size_t ws_size,
                              hipStream_t stream) {
  const float* buf       = (const float*)d_in[0];  // [B, D, P]
  const float* weight    = (const float*)d_in[1];  // [P, N]
  const float* delay_raw = (const float*)d_in[2];  // [P, N]
  float* out     = (float*)d_out;                  // [B, N]
  float* partial = (float*)d_ws;                   // [P_BLOCKS, B, N] = 4 MB

  dim3 grid(N_BLOCKS, P_BLOCKS);                   // 8 x 32 = 256 workgroups
  dsl_main_kernel<<<grid, NB, 0, stream>>>(buf, weight, delay_raw, partial);

  const int outN = B_SZ * N_SZ;
  dsl_reduce_kernel<<<(outN + 255) / 256, 256, 0, stream>>>(partial, out);
}